// RGCN_26963804684788
// MI455X (gfx1250) — compile-verified
//
#include <hip/hip_runtime.h>

#define N_NODES 100000
#define N_EDGES 1600000
#define HID 64
#define NH (N_NODES * HID)
#define LDA 68   // LDS row stride (floats): 68 % 64 = 4 -> conflict-free fragment reads

typedef __attribute__((ext_vector_type(2))) float v2f;
typedef __attribute__((ext_vector_type(8))) float v8f;

static __device__ __forceinline__ v8f wmma_f32(v2f a, v2f b, v8f c) {
    // (neg_a, A, neg_b, B, c_mod, C, reuse_a, reuse_b)
    return __builtin_amdgcn_wmma_f32_16x16x4_f32(false, a, false, b, (short)0, c,
                                                 false, false);
}

// ---- h[n] = emb[x[n]] (materialize embeddings, float4 vectorized) ----
__global__ void k_embed(const int* __restrict__ x, const float4* __restrict__ emb,
                        float4* __restrict__ h) {
    int t = blockIdx.x * blockDim.x + threadIdx.x;
    if (t >= N_NODES * 16) return;
    int n = t >> 4, c = t & 15;
    h[t] = emb[(size_t)x[n] * 16 + c];
}

// ---- edge type + per-(dst,rel) counts ----
__global__ void k_edge_type(const int* __restrict__ ei, const int* __restrict__ nt,
                            int* __restrict__ et, int* __restrict__ cnt) {
    int e = blockIdx.x * blockDim.x + threadIdx.x;
    if (e >= N_EDGES) return;
    int s = ei[e], d = ei[N_EDGES + e];
    int t = (nt[s] == nt[d]) ? 1 : 0;
    et[e] = t;
    atomicAdd(&cnt[d * 2 + t], 1);
}

__global__ void k_inv(const int* __restrict__ cnt, float* __restrict__ inv) {
    int i = blockIdx.x * blockDim.x + threadIdx.x;
    if (i >= N_NODES * 2) return;
    int c = cnt[i];
    inv[i] = 1.0f / (float)(c > 1 ? c : 1);
}

// ---- layer-1 pre-transform scatter: S_r[dst] += h[src], one wave per edge ----
__global__ void k_scatter1(const int* __restrict__ ei, const int* __restrict__ et,
                           const float* __restrict__ h,
                           float* __restrict__ S0, float* __restrict__ S1) {
    int gid = blockIdx.x * blockDim.x + threadIdx.x;
    int e = gid >> 5, lane = gid & 31;
    if (e >= N_EDGES) return;
    int s = ei[e], d = ei[N_EDGES + e];
    float2 v = *(const float2*)(h + (size_t)s * HID + lane * 2);
    float* dp = (et[e] ? S1 : S0) + (size_t)d * HID + lane * 2;
    atomicAdd(dp, v.x);
    atomicAdd(dp + 1, v.y);
}

// ---- fused layer-1 GEMMs via f32 WMMA + mean-norm + root + bias + ReLU ----
__global__ __launch_bounds__(128) void k_gemm1(
    const float* __restrict__ S0, const float* __restrict__ S1,
    const float* __restrict__ h,  const float* __restrict__ W1,
    const float* __restrict__ root1, const float* __restrict__ b1,
    const float* __restrict__ inv, float* __restrict__ h1, int nRB) {
    __shared__ float lA0[16 * LDA];
    __shared__ float lA1[16 * LDA];
    __shared__ float lAh[16 * LDA];
    __shared__ float lInv[32];

    const int tid  = threadIdx.x;
    const int lane = tid & 31;
    const int wv   = tid >> 5;        // wave id = 16-column tile
    const int m    = lane & 15;
    const int kh   = lane >> 4;       // K-half select for A/B fragments
    const int col  = wv * 16 + m;

    // Register-cache B fragments for this wave's column tile (reused all row blocks)
    v2f B0[16], B1[16], BR[16];
#pragma unroll
    for (int kk = 0; kk < 16; ++kk) {
        int k0 = kk * 4 + kh * 2;
        v2f t;
        t.x = W1[(k0) * 64 + col];        t.y = W1[(k0 + 1) * 64 + col];        B0[kk] = t;
        t.x = W1[4096 + (k0) * 64 + col]; t.y = W1[4096 + (k0 + 1) * 64 + col]; B1[kk] = t;
        t.x = root1[(k0) * 64 + col];     t.y = root1[(k0 + 1) * 64 + col];     BR[kk] = t;
    }
    const float bias = b1[col];

    for (int rb = blockIdx.x; rb < nRB; rb += gridDim.x) {
        const int rowbase = rb * 16;
        // cooperative stage of three 16x64 A tiles into LDS (coalesced float4)
        {
            const float4* s0 = (const float4*)(S0 + (size_t)rowbase * HID);
            const float4* s1 = (const float4*)(S1 + (size_t)rowbase * HID);
            const float4* sh = (const float4*)(h  + (size_t)rowbase * HID);
#pragma unroll
            for (int j = 0; j < 2; ++j) {
                int f = tid * 2 + j;            // 0..255 float4s per tile
                int r = f >> 4, c4 = f & 15;
                *(float4*)&lA0[r * LDA + c4 * 4] = s0[f];
                *(float4*)&lA1[r * LDA + c4 * 4] = s1[f];
                *(float4*)&lAh[r * LDA + c4 * 4] = sh[f];
            }
            if (tid < 32) lInv[tid] = inv[rowbase * 2 + tid];
        }
        __syncthreads();

        v8f acc0 = {}, acc1 = {}, accR = {};
#pragma unroll
        for (int kk = 0; kk < 16; ++kk) {
            int k0 = kk * 4 + kh * 2;
            v2f a0, a1, ah;
            a0.x = lA0[m * LDA + k0]; a0.y = lA0[m * LDA + k0 + 1];
            a1.x = lA1[m * LDA + k0]; a1.y = lA1[m * LDA + k0 + 1];
            ah.x = lAh[m * LDA + k0]; ah.y = lAh[m * LDA + k0 + 1];
            acc0 = wmma_f32(a0, B0[kk], acc0);
            acc1 = wmma_f32(a1, B1[kk], acc1);
            accR = wmma_f32(ah, BR[kk], accR);
        }

        // epilogue: D layout -> VGPR i holds row (i + 8*kh), col = lane&15
#pragma unroll
        for (int i = 0; i < 8; ++i) {
            int ri = i + kh * 8;
            float v = acc0[i] * lInv[2 * ri] + acc1[i] * lInv[2 * ri + 1] +
                      accR[i] + bias;
            h1[(size_t)(rowbase + ri) * HID + col] = v > 0.f ? v : 0.f;
        }
        __syncthreads();
    }
}

// ---- layer-2 per-node scalars: g_r[n] = h1[n].W2[r]; out[n] = h1[n].root2 + b2 ----
__global__ void k_g2(const float* __restrict__ h1, const float* __restrict__ W2,
                     const float* __restrict__ root2, const float* __restrict__ b2,
                     float* __restrict__ g, float* __restrict__ out) {
    int gid = blockIdx.x * blockDim.x + threadIdx.x;
    int n = gid >> 5, lane = gid & 31;
    if (n >= N_NODES) return;
    float2 hv = *(const float2*)(h1 + (size_t)n * HID + lane * 2);
    float p0 = hv.x * W2[lane * 2]        + hv.y * W2[lane * 2 + 1];
    float p1 = hv.x * W2[64 + lane * 2]   + hv.y * W2[64 + lane * 2 + 1];
    float pr = hv.x * root2[lane * 2]     + hv.y * root2[lane * 2 + 1];
#pragma unroll
    for (int off = 16; off > 0; off >>= 1) {
        p0 += __shfl_xor(p0, off, 32);
        p1 += __shfl_xor(p1, off, 32);
        pr += __shfl_xor(pr, off, 32);
    }
    if (lane == 0) {
        g[n * 2 + 0] = p0;
        g[n * 2 + 1] = p1;
        out[n] = pr + b2[0];
    }
}

// ---- layer-2 edge pass: scalar per edge (64x less traffic than row scatter) ----
__global__ void k_scatter2(const int* __restrict__ ei, const int* __restrict__ et,
                           const float* __restrict__ g, const float* __restrict__ inv,
                           float* __restrict__ out) {
    int e = blockIdx.x * blockDim.x + threadIdx.x;
    if (e >= N_EDGES) return;
    int s = ei[e], d = ei[N_EDGES + e], t = et[e];
    atomicAdd(&out[d], g[s * 2 + t] * inv[d * 2 + t]);
}

extern "C" void kernel_launch(void* const* d_in, const int* in_sizes, int n_in,
                              void* d_out, int out_size, void* d_ws, size_t ws_size,
                              hipStream_t stream) {
    const int*   x     = (const int*)  d_in[0];
    const int*   ei    = (const int*)  d_in[1];   // [2, E]: row0 = src, row1 = dst
    const int*   nt    = (const int*)  d_in[2];
    const float* emb   = (const float*)d_in[3];
    const float* W1    = (const float*)d_in[4];   // (2,64,64)
    const float* root1 = (const float*)d_in[5];   // (64,64)
    const float* b1    = (const float*)d_in[6];   // (64,)
    const float* W2    = (const float*)d_in[7];   // (2,64,1)
    const float* root2 = (const float*)d_in[8];   // (64,1)
    const float* b2    = (const float*)d_in[9];   // (1,)
    float* out = (float*)d_out;

    float* ws  = (float*)d_ws;
    float* h   = ws;                              // N*64
    float* S0  = ws + (size_t)NH;                 // N*64
    float* S1  = ws + (size_t)2 * NH;             // N*64  (S0,S1 contiguous for memset)
    float* h1  = ws + (size_t)3 * NH;             // N*64
    float* g   = ws + (size_t)4 * NH;             // N*2
    float* inv = g + 2 * N_NODES;                 // N*2
    int*   cnt = (int*)(inv + 2 * N_NODES);       // N*2
    int*   et  = cnt + 2 * N_NODES;               // E

    hipMemsetAsync(S0, 0, (size_t)2 * NH * sizeof(float), stream);
    hipMemsetAsync(cnt, 0, (size_t)2 * N_NODES * sizeof(int), stream);

    k_embed<<<(N_NODES * 16 + 255) / 256, 256, 0, stream>>>(x, (const float4*)emb,
                                                            (float4*)h);
    k_edge_type<<<(N_EDGES + 255) / 256, 256, 0, stream>>>(ei, nt, et, cnt);
    k_inv<<<(2 * N_NODES + 255) / 256, 256, 0, stream>>>(cnt, inv);
    k_scatter1<<<(N_EDGES * 32 + 255) / 256, 256, 0, stream>>>(ei, et, h, S0, S1);

    const int nRB = N_NODES / 16;                 // 6250, exact
    k_gemm1<<<1280, 128, 0, stream>>>(S0, S1, h, W1, root1, b1, inv, h1, nRB);

    k_g2<<<(N_NODES * 32 + 255) / 256, 256, 0, stream>>>(h1, W2, root2, b2, g, out);
    k_scatter2<<<(N_EDGES + 255) / 256, 256, 0, stream>>>(ei, et, g, inv, out);
}